// Correlation_27797028340332
// MI455X (gfx1250) — compile-verified
//
#include <hip/hip_runtime.h>

// ---------------------------------------------------------------------------
// FlowNetC correlation on gfx1250 via V_WMMA_F32_16X16X4_F32.
//
// out[b, dy*21+dx, y, x] = (1/256) * sum_c in1[b,c,y,x] * in2p[b,c,y+2dy,x+2dx]
// (in2p zero-padded by 20 on H and W).
//
// Parity split (stride-2 displacements): for fixed (b,y,dy,parity p),
//   out[xh, dx] = sum_c A[xh,c] * Bp[xh+dx, c]
// with A  = in1 row y, even/odd x        (64 rows x 256 ch)
//      Bp = in2 padded row y+2dy, cols   (96 rows staged, 84 meaningful)
// Banded GEMM: each wave owns TWO adjacent 16-row M-tiles; their B bands
// overlap, so 2 A-loads + 4 shared B-tile loads feed 6 WMMAs per K-step
// (1.0 LDS b64-load per WMMA -> fits the 256 B/clk LDS ceiling per WGP).
// Valid band entries (0 <= dx < 21) are extracted at store time.
// ---------------------------------------------------------------------------

typedef float v2f __attribute__((ext_vector_type(2)));
typedef float v8f __attribute__((ext_vector_type(8)));

#define CCH   256      // channels
#define HH    96       // height
#define WW    128      // width
#define DD    21       // displacements per axis
#define KC    32       // channels per LDS chunk
#define NKC   (CCH / KC)
#define AST   36       // LDS row stride in floats (32+4 pad; 36 % 32 == 4 -> conflict-free b64 reads)
#define AROWS 64       // x_half positions per parity
#define BROWS 96       // staged Bp rows per parity (rows with gx OOB are zero-filled)

#define WMMA_F32(A, B, C)                                                     \
  __builtin_amdgcn_wmma_f32_16x16x4_f32(false, (A), false, (B), (short)0,     \
                                        (C), false, false)

__global__ __launch_bounds__(128) void corr_wmma_kernel(
    const float* __restrict__ in1,
    const float* __restrict__ in2,
    float* __restrict__ out) {
  // LDS: [parity][row * AST + c]
  __shared__ float Ae[2][AROWS * AST];   // 2*64*36*4 = 18432 B
  __shared__ float Be[2][BROWS * AST];   // 2*96*36*4 = 27648 B   (total 46080 B)

  const int t    = threadIdx.x;
  const int lane = t & 31;
  const int wave = t >> 5;          // 0..3
  const int p    = wave >> 1;       // parity owned by this wave
  const int m0   = (wave & 1) * 2;  // first M-tile of this wave's pair

  const int bid = blockIdx.x;
  const int b   = bid / (HH * 3);
  const int rem = bid % (HH * 3);
  const int y   = rem / 3;
  const int dyg = rem % 3;          // dy = dyg*7 + i, i in [0,7)

  const long chanStride = (long)HH * WW;
  const long base1 = ((long)b * CCH) * chanStride + (long)y * WW;

  const int laneLo = lane & 15;     // N col / M row within a tile
  const int laneHi = lane >> 4;     // K sub-pair select (and M high half of C/D)

  // Fixed staging assignment for A: one (parity, x_half) column per thread.
  const int a_pp = t >> 6;          // 0..1
  const int a_xh = t & 63;          // 0..63

  for (int i = 0; i < 7; ++i) {
    const int dy = dyg * 7 + i;
    const int ry = y + 2 * dy - 20;              // source row in (unpadded) in2
    const bool rowok = (ry >= 0) && (ry < HH);
    const long base2 =
        ((long)b * CCH) * chanStride + (long)(rowok ? ry : 0) * WW;

    v8f acc[2][3];
    {
      const v8f z = {};
#pragma unroll
      for (int mt = 0; mt < 2; ++mt)
#pragma unroll
        for (int r = 0; r < 3; ++r) acc[mt][r] = z;
    }

    for (int kc = 0; kc < NKC; ++kc) {
      const int cbase = kc * KC;
      __syncthreads();  // WAR: previous chunk's readers done before restage

      // ---- stage A: thread walks 32 channels of its column (uniform stride)
      {
        const float* g = &in1[base1 + (long)cbase * chanStride + (2 * a_xh + a_pp)];
        float* l = &Ae[a_pp][a_xh * AST];
#pragma unroll 8
        for (int cl = 0; cl < KC; ++cl) l[cl] = g[(long)cl * chanStride];
      }
      // ---- stage Bp: 2*96 columns split into 384 half-columns of 16 ch;
      //      exactly 3 per thread (wave-balanced). OOB -> 0 (branchless).
#pragma unroll
      for (int s = 0; s < 3; ++s) {
        const int task = t + 128 * s;            // 0..383
        const int pp = (task >= 2 * BROWS) ? 1 : 0;
        const int r2 = task - 2 * BROWS * pp;    // 0..191
        const int j  = r2 >> 1;                  // 0..95
        const int kh = (r2 & 1) * 16;            // channel half
        const int gx = 2 * j + pp - 20;          // unpadded x in in2
        const bool ok = rowok && (gx >= 0) && (gx < WW);
        const int gxc = gx < 0 ? 0 : (gx > (WW - 1) ? (WW - 1) : gx);
        const float* g = &in2[base2 + (long)(cbase + kh) * chanStride + gxc];
        float* l = &Be[pp][j * AST + kh];
#pragma unroll 8
        for (int cl = 0; cl < 16; ++cl) {
          const float v = g[(long)cl * chanStride];
          l[cl] = ok ? v : 0.0f;
        }
      }
      __syncthreads();  // RAW: staging visible before compute

      // ---- 8 K-steps of 4: 2 A-tiles + 4 B-tiles -> 6 WMMAs
      const float* ap0 = &Ae[p][(m0 * 16 + laneLo) * AST + 2 * laneHi];
      const float* ap1 = ap0 + 16 * AST;
      const float* bp  = &Be[p][(m0 * 16 + laneLo) * AST + 2 * laneHi];
#pragma unroll
      for (int k = 0; k < KC; k += 4) {
        const v2f a0 = *(const v2f*)(ap0 + k);
        const v2f a1 = *(const v2f*)(ap1 + k);
        const v2f b0 = *(const v2f*)(bp + k);
        const v2f b1 = *(const v2f*)(bp + 16 * AST + k);
        const v2f b2 = *(const v2f*)(bp + 32 * AST + k);
        const v2f b3 = *(const v2f*)(bp + 48 * AST + k);
        acc[0][0] = WMMA_F32(a0, b0, acc[0][0]);
        acc[0][1] = WMMA_F32(a0, b1, acc[0][1]);
        acc[0][2] = WMMA_F32(a0, b2, acc[0][2]);
        acc[1][0] = WMMA_F32(a1, b1, acc[1][0]);
        acc[1][1] = WMMA_F32(a1, b2, acc[1][1]);
        acc[1][2] = WMMA_F32(a1, b3, acc[1][2]);
      }
    }

    // ---- extract valid band entries and store (mean over C => *1/256)
    // C/D layout: VGPR v, lane<16 -> row v, col lane; lane>=16 -> row v+8.
    // For M-tile m0+mt and B-tile (m0+mt+r): dx = 16*r + col - row.
    const float scale = 1.0f / 256.0f;
#pragma unroll
    for (int mt = 0; mt < 2; ++mt) {
#pragma unroll
      for (int v = 0; v < 8; ++v) {
        const int xloc = v + 8 * laneHi;
        const int x = 2 * ((m0 + mt) * 16 + xloc) + p;
#pragma unroll
        for (int r = 0; r < 3; ++r) {
          const int dx = 16 * r + laneLo - xloc;
          if (dx >= 0 && dx < DD) {
            const long o =
                (((long)b * (DD * DD) + (dy * DD + dx)) * HH + y) * WW + x;
            out[o] = acc[mt][r][v] * scale;
          }
        }
      }
    }
  }
}

extern "C" void kernel_launch(void* const* d_in, const int* in_sizes, int n_in,
                              void* d_out, int out_size, void* d_ws, size_t ws_size,
                              hipStream_t stream) {
  (void)n_in; (void)out_size; (void)d_ws; (void)ws_size;
  const float* in1 = (const float*)d_in[0];
  const float* in2 = (const float*)d_in[1];
  float* out = (float*)d_out;

  const int B = in_sizes[0] / (CCH * HH * WW);   // 8 for the reference shapes
  dim3 grid((unsigned)(B * HH * 3));             // (b, y, dy-group) -> 2304 blocks
  dim3 block(128);                               // 4 waves: {parity} x {M-tile pair}
  hipLaunchKernelGGL(corr_wmma_kernel, grid, block, 0, stream, in1, in2, out);
}